// PyramidRoIAlign_11742440587332
// MI455X (gfx1250) — compile-verified
//
#include <hip/hip_runtime.h>
#include <hip/hip_bf16.h>
#include <math.h>

#define POOL   7
#define NBINS  49          // 7*7
#define CCH    256         // channels
#define KMAX   16          // max patch pixels for WMMA fast path (pad K to 16)

typedef float v2f __attribute__((ext_vector_type(2)));
typedef float v8f __attribute__((ext_vector_type(8)));

// One block (256 threads = 8 wave32) per box.
// Fast path: pooled(49x256) = Wmat(49xK) @ Patch(Kx256) via V_WMMA_F32_16X16X4_F32.
// Fallback (patch > 16 px, never hit for this data): coalesced scalar bilinear.
__global__ __launch_bounds__(256) void pyramid_roi_align_kernel(
    const float* __restrict__ boxes,
    const float* __restrict__ f2, const float* __restrict__ f3,
    const float* __restrict__ f4, const float* __restrict__ f5,
    float* __restrict__ out)
{
    __shared__ float Wmat[64][KMAX];   // A: bins (padded 64) x patch pixels (padded 16)
    __shared__ float Pp[KMAX][CCH];    // B: patch pixels x channels
    __shared__ float wtab[NBINS][4];   // per-bin bilinear weights
    __shared__ int   itab[NBINS][4];   // per-bin corner indices (y*W+x)
    __shared__ int   vtab[NBINS];      // per-bin validity
    __shared__ int   bounds[4];        // ymin, ymax, xmin, xmax over valid bins

    const int box = blockIdx.x;
    const int tid = threadIdx.x;

    // ---- uniform per-box level selection (every thread recomputes; broadcast loads) ----
    const float b1 = boxes[box*5 + 1];
    const float b2 = boxes[box*5 + 2];
    const float b3 = boxes[box*5 + 3];
    const float b4 = boxes[box*5 + 4];
    const int   bidx = (int)boxes[box*5 + 0];

    const float hh = b3 - b1;
    const float ww = b4 - b2;
    // roi_level = 4 + log2(sqrt(h*w) / (224/sqrt(1024*1024)))
    const float lvlf = 4.0f + log2f(sqrtf(fmaxf(hh*ww, 1e-30f)) * (1024.0f/224.0f));
    int lvl = (int)rintf(lvlf);               // round-half-even like jnp.round
    lvl = lvl < 2 ? 2 : (lvl > 5 ? 5 : lvl);

    const int   H  = 1024 >> lvl;             // 256,128,64,32
    const int   W  = H;
    const int   shw = 10 - lvl;               // log2(W)
    const float scale = 1.0f / (float)(1 << (lvl + 2));
    const float* fm = (lvl == 2) ? f2 : (lvl == 3) ? f3 : (lvl == 4) ? f4 : f5;
    const float* plane0 = fm + (size_t)bidx * CCH * H * W;  // (batch, ch=0) base

    if (tid == 0) {
        bounds[0] = 1 << 30; bounds[1] = -(1 << 30);
        bounds[2] = 1 << 30; bounds[3] = -(1 << 30);
    }
    // zero the A matrix (rows for bins 49..63 stay zero -> zero output rows)
    for (int i = tid; i < 64 * KMAX; i += 256) ((float*)Wmat)[i] = 0.0f;
    __syncthreads();

    // ---- per-bin sample geometry (threads 0..48) ----
    if (tid < NBINS) {
        const int py = tid / POOL, px = tid % POOL;
        // NOTE reference swaps: x1=rois[:,0]=boxes[:,1], y1=rois[:,1]=boxes[:,2], ...
        const float x1f = b1 * scale, y1f = b2 * scale;
        const float x2f = b3 * scale, y2f = b4 * scale;
        const float rw  = fmaxf(x2f - x1f, 1.0f);
        const float rh  = fmaxf(y2f - y1f, 1.0f);
        const float bwf = rw * (1.0f / POOL);
        const float bhf = rh * (1.0f / POOL);
        const float y = y1f + ((float)py + 0.5f) * bhf;
        const float x = x1f + ((float)px + 0.5f) * bwf;
        const bool valid = (y > -1.0f) && (y < (float)H) && (x > -1.0f) && (x < (float)W);
        const float yc = fminf(fmaxf(y, 0.0f), (float)(H - 1));
        const float xc = fminf(fmaxf(x, 0.0f), (float)(W - 1));
        const int y0  = (int)floorf(yc);
        const int x0  = (int)floorf(xc);
        const int y1i = min(y0 + 1, H - 1);
        const int x1i = min(x0 + 1, W - 1);
        const float ly = yc - (float)y0, lx = xc - (float)x0;
        const float hy = 1.0f - ly,      hx = 1.0f - lx;
        float w00 = hy*hx, w01 = hy*lx, w10 = ly*hx, w11 = ly*lx;
        if (!valid) { w00 = w01 = w10 = w11 = 0.0f; }
        wtab[tid][0] = w00; wtab[tid][1] = w01; wtab[tid][2] = w10; wtab[tid][3] = w11;
        itab[tid][0] = y0*W + x0;  itab[tid][1] = y0*W + x1i;
        itab[tid][2] = y1i*W + x0; itab[tid][3] = y1i*W + x1i;
        vtab[tid] = valid ? 1 : 0;
        if (valid) {
            atomicMin(&bounds[0], y0);  atomicMax(&bounds[1], y1i);
            atomicMin(&bounds[2], x0);  atomicMax(&bounds[3], x1i);
        }
    }
    __syncthreads();

    int ymin = bounds[0], ymax = bounds[1], xmin = bounds[2], xmax = bounds[3];
    if (ymin > ymax) { ymin = ymax = xmin = xmax = 0; }   // all bins invalid
    const int ph = ymax - ymin + 1;
    const int pw = xmax - xmin + 1;
    const int K  = ph * pw;
    float* const outBox = out + (size_t)box * (CCH * NBINS);

    if (K <= KMAX) {
        // ==================== WMMA fast path ====================
        // Scatter bilinear weights into A (each bin owns one row; coincident
        // clamped corners accumulate).
        if (tid < NBINS && vtab[tid]) {
            const int mw = W - 1;
            #pragma unroll
            for (int j = 0; j < 4; ++j) {
                const int idx = itab[tid][j];
                const int yy = (idx >> shw) - ymin;
                const int xx = (idx & mw)   - xmin;
                Wmat[tid][yy * pw + xx] += wtab[tid][j];
            }
        }
        // Stage patch (B matrix) in LDS; thread == channel. Zero padded K rows
        // (garbage * 0 would be NaN under WMMA 0*Inf/NaN rules).
        for (int k = 0; k < K; ++k) {
            const int yy = ymin + k / pw;
            const int xx = xmin + k % pw;
            Pp[k][tid] = plane0[(size_t)tid * (H * W) + yy * W + xx];
        }
        for (int k = K; k < KMAX; ++k) Pp[k][tid] = 0.0f;
        __syncthreads();

        const int wid = tid >> 5, lane = tid & 31;
        const int lh  = lane >> 4;          // half-wave select (K/M offset)
        const int l16 = lane & 15;
        // 16 N-tiles over 8 waves (2 each), 4 M-tiles, 4 K-steps of 4.
        for (int nt = wid * 2; nt < wid * 2 + 2; ++nt) {
            const int ch = nt * 16 + l16;   // this lane's output channel column
            #pragma unroll
            for (int mt = 0; mt < 4; ++mt) {
                v8f acc = {};
                #pragma unroll
                for (int ks = 0; ks < 4; ++ks) {
                    const int kb = ks * 4 + lh * 2;   // ISA A-layout: lanes 16-31 hold K+2
                    v2f a, b;
                    a[0] = Wmat[mt * 16 + l16][kb];
                    a[1] = Wmat[mt * 16 + l16][kb + 1];
                    b[0] = Pp[kb][ch];
                    b[1] = Pp[kb + 1][ch];
                    acc = __builtin_amdgcn_wmma_f32_16x16x4_f32(
                        /*neg_a=*/false, a, /*neg_b=*/false, b,
                        /*c_mod=*/(short)0, acc, /*reuse_a=*/false, /*reuse_b=*/false);
                }
                // D layout: reg r -> M = r + 8*(lane>=16), N = lane&15
                #pragma unroll
                for (int r = 0; r < 8; ++r) {
                    const int bin = mt * 16 + r + 8 * lh;
                    if (bin < NBINS) outBox[ch * NBINS + bin] = acc[r];
                }
            }
        }
    } else {
        // ==================== general fallback (coalesced) ====================
        #pragma unroll 1
        for (int i = 0; i < NBINS; ++i) {
            const unsigned o   = (unsigned)(i * 256 + tid);   // flat (ch*49+bin)
            const unsigned c   = o / 49u;
            const unsigned bin = o - c * 49u;
            const float* pl = plane0 + (size_t)c * (H * W);
            const float v = wtab[bin][0] * pl[itab[bin][0]]
                          + wtab[bin][1] * pl[itab[bin][1]]
                          + wtab[bin][2] * pl[itab[bin][2]]
                          + wtab[bin][3] * pl[itab[bin][3]];
            outBox[o] = v;
        }
    }
}

extern "C" void kernel_launch(void* const* d_in, const int* in_sizes, int n_in,
                              void* d_out, int out_size, void* d_ws, size_t ws_size,
                              hipStream_t stream) {
    const float* boxes = (const float*)d_in[0];
    const float* f2    = (const float*)d_in[1];
    const float* f3    = (const float*)d_in[2];
    const float* f4    = (const float*)d_in[3];
    const float* f5    = (const float*)d_in[4];
    float* out = (float*)d_out;
    const int n_boxes = in_sizes[0] / 5;
    pyramid_roi_align_kernel<<<n_boxes, 256, 0, stream>>>(boxes, f2, f3, f4, f5, out);
}